// Bbox_loss2_44040594653651
// MI455X (gfx1250) — compile-verified
//
#include <hip/hip_runtime.h>
#include <hip/hip_bf16.h>

typedef float v2f __attribute__((ext_vector_type(2)));
typedef float v8f __attribute__((ext_vector_type(8)));

#define NTHREADS 768
#define NWAVES   (NTHREADS / 32)

__device__ __forceinline__ float smooth_l1(float d) {
    float ad = fabsf(d);
    return (ad < 1.0f) ? (0.5f * d * d) : (ad - 0.5f);
}

// Exact fp32 sum of the 32 lane values in this wave using one
// V_WMMA_F32_16X16X4_F32:  A = [loss, 0] per lane, B = ones.
// A layout (16x4 f32): lanes 0-15 -> (M=lane, K=0/1), lanes 16-31 -> (M=lane-16, K=2/3)
// => rowsum(m) = loss(lane m) + loss(lane m+16).
// D layout: VGPR r, lanes 0-15 -> M=r, lanes 16-31 -> M=r+8
// => sum(D[0..7]) gives rows 0-7 on lanes <16 and rows 8-15 on lanes >=16;
//    one shfl_xor(16) completes the full wave sum on every lane.
__device__ __forceinline__ float wave_sum_wmma(float v) {
    v2f a; a.x = v;    a.y = 0.0f;
    v2f b; b.x = 1.0f; b.y = 1.0f;
    v8f c = {};
    v8f d = __builtin_amdgcn_wmma_f32_16x16x4_f32(
        /*neg_a=*/false, a, /*neg_b=*/false, b,
        /*c_mod=*/(short)0, c, /*reuse_a=*/false, /*reuse_b=*/false);
    float s = d[0] + d[1] + d[2] + d[3] + d[4] + d[5] + d[6] + d[7];
    return s + __shfl_xor(s, 16, 32);
}

__global__ __launch_bounds__(NTHREADS)
void bbox_loss_kernel(const float* __restrict__ out1,
                      const float* __restrict__ out3,
                      const float* __restrict__ out5,
                      const int*   __restrict__ coord0,
                      const int*   __restrict__ coord1,
                      const int*   __restrict__ coord2,
                      const float* __restrict__ diff0,
                      const float* __restrict__ diff1,
                      const float* __restrict__ diff2,
                      float* __restrict__ d_out)
{
    __shared__ float s_loss[NWAVES];
    __shared__ float s_w[NWAVES];

    const int tid   = threadIdx.x;
    const int level = tid >> 8;        // 0..2
    const int b     = (tid >> 7) & 1;  // 0..1
    const int m     = tid & 127;       // 0..127

    const float* pred_base;
    const int*   cbase;
    const float* dbase;
    int G;
    if (level == 0)      { pred_base = out1; cbase = coord2; dbase = diff2; G = 96; }
    else if (level == 1) { pred_base = out3; cbase = coord1; dbase = diff1; G = 48; }
    else                 { pred_base = out5; cbase = coord0; dbase = diff0; G = 24; }

    // coord[b, m, 0..3]
    const int* c = cbase + (((b << 7) + m) << 2);
    int a = c[0], x = c[1], y = c[2], z = c[3];

    const bool valid  = (a > -1);
    const bool active = (cbase[b << 9] > -1);   // coord[b, 0, 0] > -1
    const bool mask   = valid && active;

    float loss = 0.0f;
    if (mask) {
        a = max(a, 0); x = max(x, 0); y = max(y, 0); z = max(z, 0);
        const float* dg = dbase + (((b << 7) + m) << 2);
        const int GG  = G * G;
        const int vol = GG * G;
        // channel = k*3 + a ; flat: ((b*12 + k*3 + a)*vol + x*GG + y*G + z)
        const int base = (b * 12 + a) * vol + x * GG + y * G + z;
        const int kstride = 3 * vol;
        float p0 = pred_base[base];
        float p1 = pred_base[base + kstride];
        float p2 = pred_base[base + 2 * kstride];
        float p3 = pred_base[base + 3 * kstride];
        loss  = smooth_l1(p0 - dg[0]);
        loss += smooth_l1(p1 - dg[1]);
        loss += smooth_l1(p2 - dg[2]);
        loss += 0.1f * smooth_l1(p3 - dg[3]);
    }

    // Per-wave reductions: loss via WMMA, weight via ballot+popcount (wave32).
    unsigned long long bal = __ballot(mask);
    float wave_w    = (float)__popcll(bal);
    float wave_loss = wave_sum_wmma(loss);

    const int wave = tid >> 5;
    const int lane = tid & 31;
    if (lane == 0) {
        s_loss[wave] = wave_loss;
        s_w[wave]    = wave_w;
    }
    __syncthreads();

    if (tid == 0) {
        float L = 0.0f, W = 0.0f;
        #pragma unroll
        for (int i = 0; i < NWAVES; ++i) { L += s_loss[i]; W += s_w[i]; }
        d_out[0] = L;   // reg_loss
        d_out[1] = W;   // reg_weight
    }
}

extern "C" void kernel_launch(void* const* d_in, const int* in_sizes, int n_in,
                              void* d_out, int out_size, void* d_ws, size_t ws_size,
                              hipStream_t stream) {
    (void)in_sizes; (void)n_in; (void)out_size; (void)d_ws; (void)ws_size;
    const float* out1   = (const float*)d_in[0];
    const float* out3   = (const float*)d_in[1];
    const float* out5   = (const float*)d_in[2];
    const int*   coord0 = (const int*)d_in[3];
    const int*   coord1 = (const int*)d_in[4];
    const int*   coord2 = (const int*)d_in[5];
    const float* diff0  = (const float*)d_in[6];
    const float* diff1  = (const float*)d_in[7];
    const float* diff2  = (const float*)d_in[8];

    bbox_loss_kernel<<<1, NTHREADS, 0, stream>>>(
        out1, out3, out5, coord0, coord1, coord2, diff0, diff1, diff2,
        (float*)d_out);
}